// Llm4SeqRec_61899068670318
// MI455X (gfx1250) — compile-verified
//
#include <hip/hip_runtime.h>
#include <hip/hip_bf16.h>
#include <math.h>

// ---------------- problem constants (from reference) ----------------
constexpr int CB    = 256;     // batch
constexpr int CS    = 200;     // seq len
constexpr int CH    = 64;      // hidden
constexpr int CK    = 7;       // num base models
constexpr int CLLM  = 768;     // llm dim
constexpr int CNITEM= 50000;

typedef float v2f __attribute__((ext_vector_type(2)));
typedef float v8f __attribute__((ext_vector_type(8)));

__device__ __forceinline__ v8f wmma_f32(v2f a, v2f b, v8f c) {
    return __builtin_amdgcn_wmma_f32_16x16x4_f32(false, a, false, b,
                                                 (short)0, c, false, false);
}

// =====================================================================
// 1) gather + mean of item_llm rows (table fits in 192MB L2)
// =====================================================================
__global__ void gather_mean_llm(const int* __restrict__ base_focus,
                                const float* __restrict__ item_llm,
                                float* __restrict__ meanllm)
{
    const int bk = blockIdx.x;          // 0 .. B*K-1
    const int t  = threadIdx.x;         // 0 .. 255
    const int* bf = base_focus + (size_t)bk * CS;
    float a0 = 0.f, a1 = 0.f, a2 = 0.f;
    for (int s = 0; s < CS; ++s) {
        int it = bf[s];
        if (it >= 0 && it < CNITEM) {
            const float* r = item_llm + (size_t)it * CLLM;
            a0 += r[t];
            a1 += r[t + 256];
            a2 += r[t + 512];
        }
    }
    const float inv = 1.0f / (float)CS;
    float* o = meanllm + (size_t)bk * CLLM;
    o[t]       = a0 * inv;
    o[t + 256] = a1 * inv;
    o[t + 512] = a2 * inv;
}

// =====================================================================
// 2) f32 WMMA GEMM, fully specialized at compile time:
//    Y[M,N] = (rowScale[m]) * A[M,KD] @ W + bias[n]
//    WTRANS=0: W stored [KD,N];  WTRANS=1: W stored [N,KD]
// =====================================================================
template<int KD, int N, bool WTRANS, bool SCALE>
__global__ void wmma_gemm(const float* __restrict__ A,
                          const float* __restrict__ W,
                          const float* __restrict__ bias,
                          const float* __restrict__ rowScale,
                          float* __restrict__ Y)
{
    const int lane = threadIdx.x & 31;
    const int half = lane >> 4;          // K-pair selector
    const int l16  = lane & 15;
    constexpr int NT = N / 16;
    const int m0 = (blockIdx.x / NT) << 4;
    const int n0 = (blockIdx.x % NT) << 4;
    const int rowA = m0 + l16;
    const int colB = n0 + l16;

    float rs = 1.0f;
    if constexpr (SCALE) rs = rowScale[rowA];

    const float* Ar = A + (size_t)rowA * KD + (half << 1);
    const float* Wp = WTRANS ? (W + (size_t)colB * KD + (half << 1))
                             : (W + (size_t)(half << 1) * N + colB);
    v8f c = {};
#pragma unroll 8
    for (int kk = 0; kk < KD; kk += 4) {
        v2f a, b;
        a.x = Ar[kk]     * rs;
        a.y = Ar[kk + 1] * rs;
        if constexpr (WTRANS) {
            b.x = Wp[kk];
            b.y = Wp[kk + 1];
        } else {
            b.x = Wp[(size_t)kk * N];
            b.y = Wp[(size_t)kk * N + N];
        }
        c = wmma_f32(a, b, c);
    }
    const float bn = bias[colB];
#pragma unroll
    for (int r = 0; r < 8; ++r) {
        Y[(size_t)(m0 + r + half * 8) * N + colB] = c[r] + bn;
    }
}

// =====================================================================
// 3) DIEN embedding gather with mask (padding idx handling)
// =====================================================================
__global__ void seq_embed(const int* __restrict__ input_seq,
                          const float* __restrict__ dien_table,
                          float* __restrict__ seq)
{
    const int gid = blockIdx.x * 256 + threadIdx.x;      // over B*S*H
    if (gid >= CB * CS * CH) return;
    const int bs = gid >> 6;
    const int hh = gid & 63;
    const int item = input_seq[bs];
    const float mask = (item != -1) ? 1.0f : 0.0f;
    const int idx = (item == -1) ? CNITEM : item;
    seq[gid] = dien_table[(size_t)idx * CH + hh] * mask;
}

// =====================================================================
// 4) GRU recurrence (torch nn.GRU). gi holds x@Wih^T+bih; Whh in LDS.
// =====================================================================
__global__ void gru_seq(const float* __restrict__ gi,    // [B,S,3H]
                        const float* __restrict__ Whh,   // [3H,H]
                        const float* __restrict__ bhh,   // [3H]
                        float* __restrict__ out)         // [B,S,H]
{
    __shared__ float sW[3 * CH * CH];   // 48 KB of the 320 KB/WGP LDS
    __shared__ float sh[CH];
    const int b = blockIdx.x;
    const int t = threadIdx.x;          // 0..63
    for (int i = t; i < 3 * CH * CH; i += 64) sW[i] = Whh[i];
    sh[t] = 0.0f;
    const float br = bhh[t], bz = bhh[t + CH], bn = bhh[t + 2 * CH];
    __syncthreads();

    const float* wr = &sW[(size_t)t * CH];
    const float* wz = &sW[(size_t)(t + CH) * CH];
    const float* wn = &sW[(size_t)(t + 2 * CH) * CH];

    for (int s = 0; s < CS; ++s) {
        float hr = br, hz = bz, hn = bn;
#pragma unroll 16
        for (int d = 0; d < CH; ++d) {
            const float hd = sh[d];
            hr += wr[d] * hd;
            hz += wz[d] * hd;
            hn += wn[d] * hd;
        }
        const float* g = gi + ((size_t)b * CS + s) * (3 * CH);
        const float r = 1.0f / (1.0f + __expf(-(g[t]          + hr)));
        const float z = 1.0f / (1.0f + __expf(-(g[t + CH]     + hz)));
        const float n = tanhf(          g[t + 2 * CH] + r * hn);
        const float h2 = (1.0f - z) * n + z * sh[t];
        __syncthreads();
        sh[t] = h2;
        out[((size_t)b * CS + s) * CH + t] = h2;
        __syncthreads();
    }
}

// =====================================================================
// 5) attention scores: scores[b,i,j] = (q_i . k_j)/8, masked.
//    Clamped unconditional loads + 0/1 mask multiply (no branches).
// =====================================================================
__global__ void attn_scores(const float* __restrict__ q,
                            const float* __restrict__ k,
                            const int* __restrict__ input_seq,
                            float* __restrict__ scores)
{
    const int b  = blockIdx.z;
    const int i0 = blockIdx.x * 16;
    const int j0 = blockIdx.y * 16;
    const int lane = threadIdx.x & 31, half = lane >> 4, l16 = lane & 15;
    const int row = i0 + l16;
    const int col = j0 + l16;
    const float rm = (row < CS) ? 1.0f : 0.0f;
    const int rowc = (row < CS) ? row : CS - 1;
    const int colc = (col < CS) ? col : CS - 1;

    const float* qr = q + ((size_t)b * CS + rowc) * CH + (half << 1);
    const float* kr = k + ((size_t)b * CS + colc) * CH + (half << 1);

    v8f c = {};
#pragma unroll
    for (int kk = 0; kk < CH; kk += 4) {
        v2f a, bf;
        a.x  = qr[kk]     * rm;          // zero out padded rows
        a.y  = qr[kk + 1] * rm;
        bf.x = kr[kk];                   // padded cols never stored
        bf.y = kr[kk + 1];
        c = wmma_f32(a, bf, c);
    }
    if (col >= CS) return;
    const float vj = (input_seq[b * CS + col] != -1) ? 1.0f : 0.0f;
#pragma unroll
    for (int r = 0; r < 8; ++r) {
        const int i = i0 + r + half * 8;
        if (i < CS) {
            const float vi = (input_seq[b * CS + i] != -1) ? 1.0f : 0.0f;
            const float val = (vi * vj != 0.0f) ? c[r] * 0.125f : -1e9f;
            scores[((size_t)b * CS + i) * CS + col] = val;
        }
    }
}

// =====================================================================
// 6) in-place softmax over last dim (len 200), one block per row
// =====================================================================
__global__ void softmax_s(float* __restrict__ scores)
{
    __shared__ float red[256];
    const size_t base = (size_t)blockIdx.x * CS;
    const int t = threadIdx.x;
    const float v = (t < CS) ? scores[base + t] : -3.4e38f;
    red[t] = v; __syncthreads();
    for (int off = 128; off; off >>= 1) {
        if (t < off) red[t] = fmaxf(red[t], red[t + off]);
        __syncthreads();
    }
    const float mx = red[0]; __syncthreads();
    const float e = (t < CS) ? __expf(v - mx) : 0.0f;
    red[t] = e; __syncthreads();
    for (int off = 128; off; off >>= 1) {
        if (t < off) red[t] += red[t + off];
        __syncthreads();
    }
    const float s = red[0];
    if (t < CS) scores[base + t] = e / s;
}

// =====================================================================
// 7) sa_raw = aw @ v per batch: [200,200]@[200,64]. 200 % 4 == 0, so the
//    WMMA K-loop needs no padding/masking at all.
// =====================================================================
__global__ void attn_av(const float* __restrict__ aw,
                        const float* __restrict__ v,
                        float* __restrict__ sa)
{
    const int b  = blockIdx.z;
    const int i0 = blockIdx.x * 16;
    const int n0 = blockIdx.y * 16;
    const int lane = threadIdx.x & 31, half = lane >> 4, l16 = lane & 15;
    const int row = i0 + l16;
    const float rm = (row < CS) ? 1.0f : 0.0f;
    const int rowc = (row < CS) ? row : CS - 1;
    const int col = n0 + l16;

    const float* ar = aw + ((size_t)b * CS + rowc) * CS + (half << 1);
    const float* vr = v  + (size_t)b * CS * CH + (size_t)(half << 1) * CH + col;

    v8f c = {};
#pragma unroll 10
    for (int kk = 0; kk < CS; kk += 4) {
        v2f a, bf;
        a.x  = ar[kk]     * rm;
        a.y  = ar[kk + 1] * rm;
        bf.x = vr[(size_t)kk * CH];
        bf.y = vr[(size_t)kk * CH + CH];
        c = wmma_f32(a, bf, c);
    }
#pragma unroll
    for (int r = 0; r < 8; ++r) {
        const int i = i0 + r + half * 8;
        if (i < CS) sa[((size_t)b * CS + i) * CH + col] = c[r];
    }
}

// =====================================================================
// 8) x = LayerNorm(sa + gru_out) per row of 64 (in-place capable)
// =====================================================================
__global__ void ln_residual(const float* __restrict__ sa,
                            const float* __restrict__ gru,
                            const float* __restrict__ g,
                            const float* __restrict__ bb,
                            float* __restrict__ x)
{
    __shared__ float red[64];
    const size_t row = blockIdx.x;
    const int t = threadIdx.x;
    const float v = sa[row * CH + t] + gru[row * CH + t];
    red[t] = v; __syncthreads();
    for (int off = 32; off; off >>= 1) { if (t < off) red[t] += red[t + off]; __syncthreads(); }
    const float m = red[0] * (1.0f / CH); __syncthreads();
    const float d = v - m;
    red[t] = d * d; __syncthreads();
    for (int off = 32; off; off >>= 1) { if (t < off) red[t] += red[t + off]; __syncthreads(); }
    const float var = red[0] * (1.0f / CH);
    x[row * CH + t] = d * rsqrtf(var + 1e-5f) * g[t] + bb[t];
}

// =====================================================================
// 9) att_w = softmax_s(x @ Wa + ba), one block per batch row
// =====================================================================
__global__ void attn_pool(const float* __restrict__ x,
                          const float* __restrict__ Wa,
                          const float* __restrict__ ba,
                          float* __restrict__ attw)
{
    __shared__ float red[256];
    const int b = blockIdx.x, t = threadIdx.x;
    float logit = -3.4e38f;
    if (t < CS) {
        float acc = ba[0];
        const float* xr = x + ((size_t)b * CS + t) * CH;
#pragma unroll 16
        for (int d = 0; d < CH; ++d) acc += xr[d] * Wa[d];
        logit = acc;
    }
    red[t] = logit; __syncthreads();
    for (int off = 128; off; off >>= 1) { if (t < off) red[t] = fmaxf(red[t], red[t + off]); __syncthreads(); }
    const float mx = red[0]; __syncthreads();
    const float e = (t < CS) ? __expf(logit - mx) : 0.0f;
    red[t] = e; __syncthreads();
    for (int off = 128; off; off >>= 1) { if (t < off) red[t] += red[t + off]; __syncthreads(); }
    if (t < CS) attw[(size_t)b * CS + t] = e / red[0];
}

// =====================================================================
// 10) preference = LN2(final[:, S-1, :] * mask) + user_table[user_id]
// =====================================================================
__global__ void pref_kernel(const float* __restrict__ fin,
                            const int* __restrict__ input_seq,
                            const int* __restrict__ user_id,
                            const float* __restrict__ ut,
                            const float* __restrict__ g,
                            const float* __restrict__ bb,
                            float* __restrict__ pref)
{
    __shared__ float red[64];
    const int b = blockIdx.x, t = threadIdx.x;
    const float mask = (input_seq[b * CS + (CS - 1)] != -1) ? 1.0f : 0.0f;
    const float v = fin[((size_t)b * CS + (CS - 1)) * CH + t] * mask;
    red[t] = v; __syncthreads();
    for (int off = 32; off; off >>= 1) { if (t < off) red[t] += red[t + off]; __syncthreads(); }
    const float m = red[0] * (1.0f / CH); __syncthreads();
    const float d = v - m;
    red[t] = d * d; __syncthreads();
    for (int off = 32; off; off >>= 1) { if (t < off) red[t] += red[t + off]; __syncthreads(); }
    const float var = red[0] * (1.0f / CH);
    const float ln = d * rsqrtf(var + 1e-5f) * g[t] + bb[t];
    pref[(size_t)b * CH + t] = ln + ut[(size_t)user_id[b] * CH + t];
}

// =====================================================================
// 11) wgts[b,k] = softmax_k( pref[b] . basemodel[b,k] )
// =====================================================================
__global__ void final_softmax(const float* __restrict__ pref,
                              const float* __restrict__ base,
                              float* __restrict__ out)
{
    __shared__ float acc[CK][CH];
    __shared__ float lg[CK];
    const int b = blockIdx.x, t = threadIdx.x;
    const float p = pref[(size_t)b * CH + t];
#pragma unroll
    for (int k = 0; k < CK; ++k)
        acc[k][t] = p * base[((size_t)b * CK + k) * CH + t];
    __syncthreads();
    if (t < CK) {
        float s = 0.0f;
        for (int d = 0; d < CH; ++d) s += acc[t][d];
        lg[t] = s;
    }
    __syncthreads();
    if (t == 0) {
        float mx = lg[0];
        for (int k = 1; k < CK; ++k) mx = fmaxf(mx, lg[k]);
        float e[CK]; float s = 0.0f;
        for (int k = 0; k < CK; ++k) { e[k] = __expf(lg[k] - mx); s += e[k]; }
        for (int k = 0; k < CK; ++k) out[(size_t)b * CK + k] = e[k] / s;
    }
}

// =====================================================================
// host-side launch. Requires d_ws >= ~152 MB.
// =====================================================================
extern "C" void kernel_launch(void* const* d_in, const int* in_sizes, int n_in,
                              void* d_out, int out_size, void* d_ws, size_t ws_size,
                              hipStream_t stream)
{
    const int*   user_id    = (const int*)d_in[0];
    const int*   input_seq  = (const int*)d_in[1];
    const int*   base_focus = (const int*)d_in[2];
    const float* user_table = (const float*)d_in[3];
    const float* item_llm   = (const float*)d_in[4];
    const float* W_proj     = (const float*)d_in[5];
    const float* b_proj     = (const float*)d_in[6];
    const float* dien_table = (const float*)d_in[7];
    const float* gru_Wih    = (const float*)d_in[8];
    const float* gru_Whh    = (const float*)d_in[9];
    const float* gru_bih    = (const float*)d_in[10];
    const float* gru_bhh    = (const float*)d_in[11];
    const float* Wq = (const float*)d_in[12]; const float* bq = (const float*)d_in[13];
    const float* Wk = (const float*)d_in[14]; const float* bk = (const float*)d_in[15];
    const float* Wv = (const float*)d_in[16]; const float* bv = (const float*)d_in[17];
    const float* Wo = (const float*)d_in[18]; const float* bo = (const float*)d_in[19];
    const float* Wa = (const float*)d_in[20]; const float* ba = (const float*)d_in[21];
    const float* ln1_g = (const float*)d_in[22]; const float* ln1_b = (const float*)d_in[23];
    const float* aug_Wih = (const float*)d_in[24]; const float* aug_Whh = (const float*)d_in[25];
    const float* aug_bih = (const float*)d_in[26]; const float* aug_bhh = (const float*)d_in[27];
    const float* ln2_g = (const float*)d_in[28]; const float* ln2_b = (const float*)d_in[29];

    float* ws = (float*)d_ws;
    // workspace layout (floats)
    const size_t o_seq  = 0;                      // B*S*H   (seq_emb, later sa_raw)
    const size_t o_gi   = o_seq  + 3276800;       // B*S*3H  (gi for both GRUs)
    const size_t o_gru  = o_gi   + 9830400;       // B*S*H   (gru_out, later final2)
    const size_t o_q    = o_gru  + 3276800;       // B*S*H   (q, later sa -> x)
    const size_t o_k    = o_q    + 3276800;       // B*S*H   (k)
    const size_t o_v    = o_k    + 3276800;       // B*S*H   (v)
    const size_t o_sc   = o_v    + 3276800;       // B*S*S   (scores/aw)
    const size_t o_aw   = o_sc   + 10240000;      // B*S     (att_w)
    const size_t o_mean = o_aw   + 51200;         // B*K*768 (mean llm emb)
    const size_t o_base = o_mean + 1376256;       // B*K*H   (basemodel)
    const size_t o_pref = o_base + 114688;        // B*H     (preference)

    const int MBS = CB * CS;                      // 51200 rows

    // --- base-model branch (independent of DIEN branch) ---
    gather_mean_llm<<<CB * CK, 256, 0, stream>>>(base_focus, item_llm, ws + o_mean);
    wmma_gemm<CLLM, CH, false, false><<<(CB * CK / 16) * (CH / 16), 32, 0, stream>>>(
        ws + o_mean, W_proj, b_proj, nullptr, ws + o_base);

    // --- DIEN branch ---
    seq_embed<<<(MBS * CH + 255) / 256, 256, 0, stream>>>(input_seq, dien_table, ws + o_seq);
    wmma_gemm<CH, 3 * CH, true, false><<<(MBS / 16) * (3 * CH / 16), 32, 0, stream>>>(
        ws + o_seq, gru_Wih, gru_bih, nullptr, ws + o_gi);
    gru_seq<<<CB, 64, 0, stream>>>(ws + o_gi, gru_Whh, gru_bhh, ws + o_gru);

    // q / k / v projections
    wmma_gemm<CH, CH, false, false><<<(MBS / 16) * (CH / 16), 32, 0, stream>>>(
        ws + o_gru, Wq, bq, nullptr, ws + o_q);
    wmma_gemm<CH, CH, false, false><<<(MBS / 16) * (CH / 16), 32, 0, stream>>>(
        ws + o_gru, Wk, bk, nullptr, ws + o_k);
    wmma_gemm<CH, CH, false, false><<<(MBS / 16) * (CH / 16), 32, 0, stream>>>(
        ws + o_gru, Wv, bv, nullptr, ws + o_v);

    // self-attention
    attn_scores<<<dim3(13, 13, CB), 32, 0, stream>>>(ws + o_q, ws + o_k, input_seq, ws + o_sc);
    softmax_s<<<MBS, 256, 0, stream>>>(ws + o_sc);
    attn_av<<<dim3(13, CH / 16, CB), 32, 0, stream>>>(ws + o_sc, ws + o_v, ws + o_seq);
    wmma_gemm<CH, CH, false, false><<<(MBS / 16) * (CH / 16), 32, 0, stream>>>(
        ws + o_seq, Wo, bo, nullptr, ws + o_q);

    // x = LN1(sa + gru_out), in place over sa
    ln_residual<<<MBS, 64, 0, stream>>>(ws + o_q, ws + o_gru, ln1_g, ln1_b, ws + o_q);

    // attention pooling weights + second GRU (rowScale folds x*att_w into GEMM)
    attn_pool<<<CB, 256, 0, stream>>>(ws + o_q, Wa, ba, ws + o_aw);
    wmma_gemm<CH, 3 * CH, true, true><<<(MBS / 16) * (3 * CH / 16), 32, 0, stream>>>(
        ws + o_q, aug_Wih, aug_bih, ws + o_aw, ws + o_gi);
    gru_seq<<<CB, 64, 0, stream>>>(ws + o_gi, aug_Whh, aug_bhh, ws + o_gru);

    // preference and final mixture weights
    pref_kernel<<<CB, 64, 0, stream>>>(ws + o_gru, input_seq, user_id, user_table,
                                       ln2_g, ln2_b, ws + o_pref);
    final_softmax<<<CB, 64, 0, stream>>>(ws + o_pref, ws + o_base, (float*)d_out);
}